// Compressor_47699906789380
// MI455X (gfx1250) — compile-verified
//
#include <hip/hip_runtime.h>
#include <math.h>

#define T_TOK 4096   // B*S
#define D_DIM 1024
#define N_EXP 64
#define R_DIM 64
#define TOPK  2
#define KC    64             // K-chunk staged in LDS for expert GEMM
#define NCHUNK (D_DIM / KC)  // 16

typedef __attribute__((ext_vector_type(2))) float v2f;
typedef __attribute__((ext_vector_type(8))) float v8f;

// Async global->LDS copy (CDNA5, tracked by ASYNCcnt).
// lds_addr: LDS byte address (low 32 bits of the generic/flat shared pointer);
// gaddr: 64-bit global address.
__device__ __forceinline__ void async_b128(unsigned lds_addr, const float4* gaddr) {
    asm volatile("global_load_async_to_lds_b128 %0, %1, off"
                 :: "v"(lds_addr), "v"(gaddr)
                 : "memory");
}

// ---------------- kernel 0: zero output region + counters ----------------
__global__ void k_zero(float* __restrict__ out, int* __restrict__ cnts) {
    int gid = blockIdx.x * blockDim.x + threadIdx.x;
    if (gid < T_TOK * R_DIM) out[gid] = 0.0f;
    if (gid < 3 * N_EXP)     cnts[gid] = 0;   // counts | offsets | cursors
}

// ---------------- kernel 1: router scores via f32 WMMA ----------------
// scores[t, n] = sum_d x[t,d] * router_w[n,d]
// grid = 256 token tiles, block = 128 threads = 4 waves (one n-tile each)
__global__ void __launch_bounds__(128) k_router(const float* __restrict__ x,
                                                const float* __restrict__ rw,
                                                float* __restrict__ scores) {
    const int tile = blockIdx.x;           // token tile (16 rows)
    const int wave = threadIdx.x >> 5;     // n-tile 0..3
    const int lane = threadIdx.x & 31;
    const int row  = lane & 15;
    const int klo  = (lane >> 4) << 1;     // lanes 16-31 hold K+2,K+3
    const int n    = row + wave * 16;      // B column for this lane

    const float* aptr = x  + (size_t)(tile * 16 + row) * D_DIM + klo;
    const float* bptr = rw + (size_t)n * D_DIM + klo;

    v8f acc = {};
#pragma unroll 4
    for (int k = 0; k < D_DIM; k += 4) {
        v2f a = *(const v2f*)(aptr + k);
        v2f b = *(const v2f*)(bptr + k);
        acc = __builtin_amdgcn_wmma_f32_16x16x4_f32(false, a, false, b,
                                                    (short)0, acc, false, false);
    }
    const int mhi = (lane >> 4) * 8;
#pragma unroll
    for (int r = 0; r < 8; ++r) {
        int m = r + mhi;
        scores[(size_t)(tile * 16 + m) * N_EXP + n] = acc[r];
    }
}

// ---------------- kernel 2: top-2 + softmax + per-expert counts ----------------
__global__ void k_top2(const float* __restrict__ scores,
                       float* __restrict__ oidx, float* __restrict__ ow,
                       int* __restrict__ cnts) {
    int t = blockIdx.x * blockDim.x + threadIdx.x;
    if (t >= T_TOK) return;
    const float* s = scores + (size_t)t * N_EXP;
    float m1 = -INFINITY, m2 = -INFINITY;
    int   i1 = 0, i2 = 0;
    for (int n = 0; n < N_EXP; ++n) {
        float v = s[n];
        if (v > m1)      { m2 = m1; i2 = i1; m1 = v; i1 = n; }
        else if (v > m2) { m2 = v;  i2 = n; }
    }
    float p   = __expf(m2 - m1);
    float inv = 1.0f / (1.0f + p);
    oidx[t * 2 + 0] = (float)i1;
    oidx[t * 2 + 1] = (float)i2;
    ow[t * 2 + 0] = inv;
    ow[t * 2 + 1] = p * inv;
    atomicAdd(&cnts[i1], 1);
    atomicAdd(&cnts[i2], 1);
}

// ---------------- kernel 3: tiny exclusive scan over 64 experts ----------------
__global__ void k_scan(const int* __restrict__ cnts, int* __restrict__ offs,
                       int* __restrict__ curs) {
    if (threadIdx.x == 0 && blockIdx.x == 0) {
        int run = 0;
        for (int e = 0; e < N_EXP; ++e) {
            offs[e] = run;
            curs[e] = run;
            run += cnts[e];
        }
    }
}

// ---------------- kernel 4: build expert-major assignment lists ----------------
__global__ void k_fill(const float* __restrict__ oidx, int* __restrict__ curs,
                       int* __restrict__ list) {
    int t = blockIdx.x * blockDim.x + threadIdx.x;
    if (t >= T_TOK) return;
#pragma unroll
    for (int r = 0; r < TOPK; ++r) {
        int e = (int)oidx[t * 2 + r];
        int slot = atomicAdd(&curs[e], 1);
        list[slot] = t * 2 + r;            // token*2 | rank  (also indexes ow[])
    }
}

// ---------------- kernel 5: grouped expert GEMM (WMMA) + weighted scatter -------
// One block per expert; 8 waves, each owning a 16-token tile of this expert's
// list, computing a 16x64 output tile over K=1024. B (neurons[e]) is staged in
// LDS with double-buffered async global->LDS copies (ASYNCcnt pipelined).
__global__ void __launch_bounds__(256) k_expert(const float* __restrict__ x,
                                                const float* __restrict__ neurons,
                                                const float* __restrict__ ow,
                                                const int* __restrict__ cnts,
                                                const int* __restrict__ offs,
                                                const int* __restrict__ list,
                                                float* __restrict__ out) {
    __shared__ float Bs[2][KC * R_DIM];    // 2 x 16 KB double buffer
    __shared__ int   toksS[8 * 16];
    __shared__ float wtsS[8 * 16];

    const int e    = blockIdx.x;
    const int cnt  = cnts[e];
    const int off  = offs[e];
    const int ntiles = (cnt + 15) >> 4;
    const int wave = threadIdx.x >> 5;
    const int lane = threadIdx.x & 31;
    const int row  = lane & 15;
    const int klo  = (lane >> 4) << 1;
    const float* Bg = neurons + (size_t)e * (D_DIM * R_DIM);

    // LDS byte addresses of the two B buffers (low 32 bits of flat address)
    const unsigned ldsB0 = (unsigned)(size_t)(const void*)&Bs[0][0];
    const unsigned ldsB1 = (unsigned)(size_t)(const void*)&Bs[1][0];

    for (int tg = 0; tg * 8 < ntiles; ++tg) {
        const int  tile   = tg * 8 + wave;
        const bool active = tile < ntiles;

        // kick off async staging of B chunk 0 into buffer 0
        // (each wave issues exactly 4 async b128 ops per chunk)
        {
            const float4* src = (const float4*)Bg;
#pragma unroll
            for (int i = 0; i < 4; ++i)
                async_b128(ldsB0 + (threadIdx.x + i * 256) * 16,
                           src + threadIdx.x + i * 256);
        }

        // stage this wave's 16 list entries (token*2|rank) + weights
        if (lane < 16) {
            int idx = tile * 16 + lane;
            int t2 = 0; float w = 0.0f;
            if (active && idx < cnt) { t2 = list[off + idx]; w = ow[t2]; }
            toksS[wave * 16 + lane] = t2;
            wtsS [wave * 16 + lane] = w;
        }
        __syncthreads();

        const int myTok = toksS[wave * 16 + row] >> 1;  // A row token for this lane
        const float* aptr = x + (size_t)myTok * D_DIM + klo;

        v8f acc0 = {}, acc1 = {}, acc2 = {}, acc3 = {};

        for (int c = 0; c < NCHUNK; ++c) {
            const int buf = c & 1;
            if (c + 1 < NCHUNK) {
                // prefetch next chunk into the alternate buffer (its readers
                // finished at the trailing barrier of the previous iteration)
                const float4* src = (const float4*)(Bg + (size_t)(c + 1) * KC * R_DIM);
                const unsigned dstb = buf ? ldsB0 : ldsB1;
#pragma unroll
                for (int i = 0; i < 4; ++i)
                    async_b128(dstb + (threadIdx.x + i * 256) * 16,
                               src + threadIdx.x + i * 256);
                // 4 newly-issued ops may stay in flight; the current chunk's 4
                // (older, in-order completion) must have landed
                asm volatile("s_wait_asynccnt 0x4" ::: "memory");
            } else {
                asm volatile("s_wait_asynccnt 0x0" ::: "memory");
            }
            __syncthreads();

            const float* Bsc  = &Bs[buf][0];
            const float* aptc = aptr + c * KC;
#pragma unroll 4
            for (int kk = 0; kk < KC; kk += 4) {
                v2f a = *(const v2f*)(aptc + kk);
                const float* brow = Bsc + (kk + klo) * R_DIM;  // +R_DIM = next K
                v2f b0, b1, b2, b3;
                b0.x = brow[row +  0]; b0.y = brow[R_DIM + row +  0];
                b1.x = brow[row + 16]; b1.y = brow[R_DIM + row + 16];
                b2.x = brow[row + 32]; b2.y = brow[R_DIM + row + 32];
                b3.x = brow[row + 48]; b3.y = brow[R_DIM + row + 48];
                acc0 = __builtin_amdgcn_wmma_f32_16x16x4_f32(false, a, false, b0,
                                                             (short)0, acc0, false, false);
                acc1 = __builtin_amdgcn_wmma_f32_16x16x4_f32(false, a, false, b1,
                                                             (short)0, acc1, false, false);
                acc2 = __builtin_amdgcn_wmma_f32_16x16x4_f32(false, a, false, b2,
                                                             (short)0, acc2, false, false);
                acc3 = __builtin_amdgcn_wmma_f32_16x16x4_f32(false, a, false, b3,
                                                             (short)0, acc3, false, false);
            }
            __syncthreads();
        }

        // weighted scatter: out[token, n] += w * acc   (exactly 2 adds per element
        // across the whole grid -> commutative, bitwise deterministic)
        const int mhi = (lane >> 4) * 8;
#pragma unroll
        for (int r = 0; r < 8; ++r) {
            int m = r + mhi;
            bool ok = active && (tile * 16 + m) < cnt;
            if (ok) {
                int   tok = toksS[wave * 16 + m] >> 1;
                float w   = wtsS [wave * 16 + m];
                float* o  = out + (size_t)tok * R_DIM;
                unsafeAtomicAdd(o + row +  0, w * acc0[r]);
                unsafeAtomicAdd(o + row + 16, w * acc1[r]);
                unsafeAtomicAdd(o + row + 32, w * acc2[r]);
                unsafeAtomicAdd(o + row + 48, w * acc3[r]);
            }
        }
        __syncthreads();
    }
}

// ---------------- host launcher ----------------
extern "C" void kernel_launch(void* const* d_in, const int* in_sizes, int n_in,
                              void* d_out, int out_size, void* d_ws, size_t ws_size,
                              hipStream_t stream) {
    (void)in_sizes; (void)n_in; (void)out_size; (void)ws_size;
    const float* x  = (const float*)d_in[0];   // [4096, 1024]
    const float* rw = (const float*)d_in[1];   // [64, 1024]
    const float* nr = (const float*)d_in[2];   // [64, 1024, 64]

    float* out  = (float*)d_out;                           // [4096, 64]
    float* oidx = out + (size_t)T_TOK * R_DIM;             // [4096, 2] (as float)
    float* ow   = oidx + (size_t)T_TOK * TOPK;             // [4096, 2]

    float* scores = (float*)d_ws;                                          // [4096,64]
    int*   cnts   = (int*)((char*)d_ws + (size_t)T_TOK * N_EXP * sizeof(float));
    int*   offs   = cnts + N_EXP;
    int*   curs   = offs + N_EXP;
    int*   list   = curs + N_EXP;                                          // [8192]

    k_zero  <<<dim3((T_TOK * R_DIM + 255) / 256), dim3(256), 0, stream>>>(out, cnts);
    k_router<<<dim3(T_TOK / 16),  dim3(128), 0, stream>>>(x, rw, scores);
    k_top2  <<<dim3(T_TOK / 256), dim3(256), 0, stream>>>(scores, oidx, ow, cnts);
    k_scan  <<<dim3(1),           dim3(1),   0, stream>>>(cnts, offs, curs);
    k_fill  <<<dim3(T_TOK / 256), dim3(256), 0, stream>>>(oidx, curs, list);
    k_expert<<<dim3(N_EXP),       dim3(256), 0, stream>>>(x, nr, ow, cnts, offs, list, out);
}